// SleepStageLoss_53730040873040
// MI455X (gfx1250) — compile-verified
//
#include <hip/hip_runtime.h>

#ifndef __has_builtin
#define __has_builtin(x) 0
#endif

#if __has_builtin(__builtin_amdgcn_global_load_async_to_lds_b128) && \
    __has_builtin(__builtin_amdgcn_s_wait_asynccnt)
#define USE_ASYNC_LDS 1
#else
#define USE_ASYNC_LDS 0
#endif

#define AS1 __attribute__((address_space(1)))
#define AS3 __attribute__((address_space(3)))

typedef int v4i __attribute__((ext_vector_type(4)));

static constexpr int kN       = 4096 * 1024;   // samples
static constexpr int kC       = 5;             // classes
static constexpr int kNights  = 4096;
static constexpr int kBlk     = 256;
static constexpr int kNBlk    = kN / kBlk;     // 16384
static constexpr int kTotalF  = kN * kC;       // 20,971,520 floats
// VALID[s1][s2] packed: bit (s1*5+s2). rows: {0,1},{0,1,2},{1,2,3},{2,3,4},{2,3,4}
static constexpr unsigned kValidPack =
    (3u << 0) | (7u << 5) | (14u << 10) | (28u << 15) | (28u << 20);

// Workspace layout (floats):
//   [0,4096)              night pair-value sums   (atomics, zeroed each call)
//   [4096,8192)           night pair counts       (atomics, zeroed each call)
//   [8192, 8192+16384)    per-block  sum(w_t)
//   [+16384, +32768)      per-block  sum(w_t*nll)
//   [+32768, +49152)      per-block  sum(smooth)

__device__ __forceinline__ float wave_sum(float v) {
#pragma unroll
  for (int off = 16; off > 0; off >>= 1) v += __shfl_down(v, off, 32);
  return v;
}

__global__ void zero_ws(float* __restrict__ ws) {
  const int i = blockIdx.x * blockDim.x + threadIdx.x;
  if (i < 2 * kNights) ws[i] = 0.0f;
}

__global__ __launch_bounds__(kBlk) void sleep_loss_main(
    const float* __restrict__ outputs,
    const int*   __restrict__ targets,
    const int*   __restrict__ nights,
    const float* __restrict__ cw,
    float*       __restrict__ ws) {
  __shared__ __align__(16) float s_out[1288];          // 257 samples * 5 + pad
  __shared__ float s_probs[(kBlk + 1) * kC];           // 1285
  __shared__ float s_red[3 * 8];

  const int tid   = threadIdx.x;
  const int base  = blockIdx.x * kBlk;
  const int base5 = base * kC;
  const int avail = kTotalF - base5;                   // floats legal to touch
  const float* gsrc = outputs + base5;

  // ---- Stage outputs tile (own 256 samples + 1 halo sample) into LDS ----
#if USE_ASYNC_LDS
#pragma unroll
  for (int k = 0; k < 2; ++k) {
    const int i4 = tid + k * kBlk;                     // 128-bit transfer index
    if (i4 < 322 && (i4 * 4 + 4) <= avail) {
      __builtin_amdgcn_global_load_async_to_lds_b128(
          (AS1 v4i*)(gsrc + i4 * 4),
          (AS3 v4i*)(s_out + i4 * 4), 0, 0);
    }
  }
  __builtin_amdgcn_s_wait_asynccnt(0);
  __syncthreads();
#else
  for (int f = tid; f < 1288 && f < avail; f += kBlk) s_out[f] = gsrc[f];
  __syncthreads();
#endif

  // ---- Per-sample softmax / log-sum-exp from LDS (stride-5: conflict-free) ----
  float x[kC];
#pragma unroll
  for (int c = 0; c < kC; ++c) x[c] = s_out[tid * kC + c];
  float m = fmaxf(fmaxf(fmaxf(x[0], x[1]), fmaxf(x[2], x[3])), x[4]);
  float e[kC], sum = 0.0f;
#pragma unroll
  for (int c = 0; c < kC; ++c) { e[c] = __expf(x[c] - m); sum += e[c]; }
  const float inv = 1.0f / sum;
  const float lse = m + __logf(sum);
#pragma unroll
  for (int c = 0; c < kC; ++c) s_probs[tid * kC + c] = e[c] * inv;

  // Halo sample (first sample of next block) -> probs slot 256
  if (tid == 0 && (base + kBlk) < kN) {
    float hm = s_out[kBlk * kC];
#pragma unroll
    for (int c = 1; c < kC; ++c) hm = fmaxf(hm, s_out[kBlk * kC + c]);
    float hs = 0.0f, he[kC];
#pragma unroll
    for (int c = 0; c < kC; ++c) { he[c] = __expf(s_out[kBlk * kC + c] - hm); hs += he[c]; }
    const float hinv = 1.0f / hs;
#pragma unroll
    for (int c = 0; c < kC; ++c) s_probs[kBlk * kC + c] = he[c] * hinv;
  }

  // ---- CE terms for own sample (no barrier needed yet) ----
  const int gi = base + tid;
  const int t0 = targets[gi];
  const float w0 = cw[0], w1 = cw[1], w2 = cw[2], w3 = cw[3], w4 = cw[4];
  const float wsumc = w0 + w1 + w2 + w3 + w4;
  const float wt = (t0 == 0) ? w0 : (t0 == 1) ? w1 : (t0 == 2) ? w2
                 : (t0 == 3) ? w3 : w4;
  const float xt = s_out[tid * kC + t0];
  const float dotwx = w0 * x[0] + w1 * x[1] + w2 * x[2] + w3 * x[3] + w4 * x[4];
  const float my_w   = wt;
  const float my_nll = wt * (lse - xt);
  const float my_sm  = wsumc * lse - dotwx;

  __syncthreads();   // s_probs (incl. halo) ready

  // ---- Pair term: |p[i+1]-p[i]|_1 * trans_w, only when same night ----
  const int n0 = nights[gi];
  float pv = 0.0f, pc = 0.0f;
  if (gi < kN - 1) {
    const int n1 = nights[gi + 1];
    if (n1 == n0) {
      const int t1 = targets[gi + 1];
      float d = 0.0f;
#pragma unroll
      for (int c = 0; c < kC; ++c)
        d += fabsf(s_probs[(tid + 1) * kC + c] - s_probs[tid * kC + c]);
      const float tw = ((kValidPack >> (t0 * 5 + t1)) & 1u) ? 1.0f : 2.0f;
      pv = d * tw;
      pc = 1.0f;
    }
  }

  // ---- Segmented wave32 reduction keyed by (sorted) night id ----
  const int lane = tid & 31;
  int key = n0;
#pragma unroll
  for (int off = 1; off < 32; off <<= 1) {
    const float v2 = __shfl_down(pv, off, 32);
    const float c2 = __shfl_down(pc, off, 32);
    const int   k2 = __shfl_down(key, off, 32);
    if ((lane + off) < 32 && k2 == key) { pv += v2; pc += c2; }
  }
  const int prev = __shfl_up(key, 1, 32);
  if ((lane == 0 || prev != key) && (pc != 0.0f || pv != 0.0f)) {
    atomicAdd(ws + key, pv);
    atomicAdd(ws + kNights + key, pc);
  }

  // ---- CE block reduction -> per-block partials (deterministic stores) ----
  const float rw = wave_sum(my_w);
  const float rn = wave_sum(my_nll);
  const float rs = wave_sum(my_sm);
  const int wv = tid >> 5;
  if (lane == 0) { s_red[wv] = rw; s_red[8 + wv] = rn; s_red[16 + wv] = rs; }
  __syncthreads();
  if (tid == 0) {
    float aw = 0.0f, an = 0.0f, am = 0.0f;
#pragma unroll
    for (int k = 0; k < 8; ++k) { aw += s_red[k]; an += s_red[8 + k]; am += s_red[16 + k]; }
    float* pw = ws + 2 * kNights;
    pw[blockIdx.x]             = aw;
    pw[kNBlk + blockIdx.x]     = an;
    pw[2 * kNBlk + blockIdx.x] = am;
  }
}

__global__ __launch_bounds__(256) void sleep_loss_final(
    const float* __restrict__ ws, float* __restrict__ out) {
  __shared__ float s_red[4 * 8];
  const int tid = threadIdx.x;

  float tsum = 0.0f;
  for (int j = tid; j < kNights; j += 256) {
    const float c = ws[kNights + j];
    tsum += (c > 0.0f) ? ws[j] / (c * (float)kC) : 0.0f;
  }
  const float* pw = ws + 2 * kNights;
  float sw = 0.0f, sn = 0.0f, sm = 0.0f;
  for (int j = tid; j < kNBlk; j += 256) {
    sw += pw[j];
    sn += pw[kNBlk + j];
    sm += pw[2 * kNBlk + j];
  }

  tsum = wave_sum(tsum);
  sw   = wave_sum(sw);
  sn   = wave_sum(sn);
  sm   = wave_sum(sm);
  const int lane = tid & 31, wv = tid >> 5;
  if (lane == 0) {
    s_red[wv] = tsum; s_red[8 + wv] = sw; s_red[16 + wv] = sn; s_red[24 + wv] = sm;
  }
  __syncthreads();
  if (tid == 0) {
    float T = 0.0f, W = 0.0f, Nl = 0.0f, S = 0.0f;
#pragma unroll
    for (int k = 0; k < 8; ++k) {
      T += s_red[k]; W += s_red[8 + k]; Nl += s_red[16 + k]; S += s_red[24 + k];
    }
    const float ce = 0.9f * (Nl / W) + 0.02f * (S / W);   // (1-0.1), 0.1/5
    out[0] = ce + 0.2f * T;                               // TEMPORAL_WEIGHT
  }
}

extern "C" void kernel_launch(void* const* d_in, const int* in_sizes, int n_in,
                              void* d_out, int out_size, void* d_ws, size_t ws_size,
                              hipStream_t stream) {
  (void)in_sizes; (void)n_in; (void)out_size; (void)ws_size;
  const float* outputs = (const float*)d_in[0];
  const int*   targets = (const int*)d_in[1];
  const int*   nights  = (const int*)d_in[2];
  const float* cw      = (const float*)d_in[3];
  float* ws  = (float*)d_ws;   // needs 2*4096 + 3*16384 floats = 224 KiB
  float* out = (float*)d_out;

  zero_ws<<<(2 * kNights + 255) / 256, 256, 0, stream>>>(ws);
  sleep_loss_main<<<kNBlk, kBlk, 0, stream>>>(outputs, targets, nights, cw, ws);
  sleep_loss_final<<<1, 256, 0, stream>>>(ws, out);
}